// lveg_65566970741220
// MI455X (gfx1250) — compile-verified
//
#include <hip/hip_runtime.h>

// ---------------------------------------------------------------------------
// Native vector types (clang ext vectors: valid for nontemporal builtins)
// ---------------------------------------------------------------------------
typedef __attribute__((ext_vector_type(4))) float v4f;
typedef __attribute__((ext_vector_type(4))) int   v4i;

// ---------------------------------------------------------------------------
// Fast device math (single-instruction V_EXP_F32 / V_LOG_F32 / V_RCP_F32)
// ---------------------------------------------------------------------------
static __device__ __forceinline__ float dexp2f(float x) { return __builtin_amdgcn_exp2f(x); }
static __device__ __forceinline__ float dlog2f(float x) { return __builtin_amdgcn_logf(x); }
static __device__ __forceinline__ float drcpf (float x) { return __builtin_amdgcn_rcpf(x); }

#define F_LOG2PI   1.83787706640934534f   /* ln(2*pi)                         */
#define F_LN2      0.69314718055994531f   /* ln(2)                            */
#define F_2LOG2E   2.88539008177792681f   /* 2*log2(e): exp(2x)=exp2(x*this)  */

// ---------------------------------------------------------------------------
// CDNA5 async global->LDS copy (ASYNCcnt path), with portable fallback
// ---------------------------------------------------------------------------
#if defined(__HIP_DEVICE_COMPILE__) && __has_builtin(__builtin_amdgcn_global_load_async_to_lds_b128)
#define ASYNC_LDS 1
#else
#define ASYNC_LDS 0
#endif

#if ASYNC_LDS
typedef __attribute__((address_space(1))) v4i* as1_v4i_p;
typedef __attribute__((address_space(3))) v4i* as3_v4i_p;
#endif

static __device__ __forceinline__ void lds_copy16(const float* g, float* l)
{
#if ASYNC_LDS
    __builtin_amdgcn_global_load_async_to_lds_b128(
        (as1_v4i_p)(const_cast<float*>(g)),
        (as3_v4i_p)(l),
        0, 0);
#else
    *(v4f*)l = *(const v4f*)g;
#endif
}

static __device__ __forceinline__ void lds_copy_wait()
{
#if ASYNC_LDS
#if __has_builtin(__builtin_amdgcn_s_wait_asynccnt)
    __builtin_amdgcn_s_wait_asynccnt(0);
#else
    asm volatile("s_wait_asynccnt 0" ::: "memory");
#endif
#endif
}

// ---------------------------------------------------------------------------
// Kernel A: cs_mu/cs_var[b,l,i,j] = gaussian_multi(s[b,l,j], trans_c[i,j])
//           + broadcast copies t_p_mu_e / t_p_var_e (NT stores)
// One block per (b,l); thread t: i = t>>3, j4 = (t&7)*4 (v4f over j).
// ---------------------------------------------------------------------------
extern "C" __global__ void __launch_bounds__(256)
lveg_cs_kernel(const int*   __restrict__ tokens,
               const float* __restrict__ tp_mu,  const float* __restrict__ tp_var,
               const float* __restrict__ tc_mu,  const float* __restrict__ tc_var,
               const float* __restrict__ s_mu_t, const float* __restrict__ s_var_t,
               float* __restrict__ cs_mu,  float* __restrict__ cs_var,
               float* __restrict__ tpmu_e, float* __restrict__ tpvar_e)
{
    __shared__ __align__(16) float s_tcmu[1024];
    __shared__ __align__(16) float s_tcvar[1024];
    __shared__ __align__(16) float s_v2c[1024];   // exp(2*tc_var)

    const int tid = threadIdx.x;
    const int bl  = blockIdx.x;          // b*128 + l
    const int i   = tid >> 3;
    const int j4  = (tid & 7) << 2;

    lds_copy16(tc_mu + tid * 4, s_tcmu + tid * 4);

    v4f tv = ((const v4f*)tc_var)[tid];
    ((v4f*)s_tcvar)[tid] = tv;
    v4f w;
    w.x = dexp2f(tv.x * F_2LOG2E);
    w.y = dexp2f(tv.y * F_2LOG2E);
    w.z = dexp2f(tv.z * F_2LOG2E);
    w.w = dexp2f(tv.w * F_2LOG2E);
    ((v4f*)s_v2c)[tid] = w;

    const int tok = tokens[bl];
    v4f smu  = *(const v4f*)(s_mu_t  + (size_t)tok * 32 + j4);
    v4f svar = *(const v4f*)(s_var_t + (size_t)tok * 32 + j4);

    // Broadcast outputs: pure table replication, stream past caches.
    v4f pm = ((const v4f*)tp_mu)[tid];
    v4f pv = ((const v4f*)tp_var)[tid];
    __builtin_nontemporal_store(pm, (v4f*)tpmu_e  + (size_t)bl * 256 + tid);
    __builtin_nontemporal_store(pv, (v4f*)tpvar_e + (size_t)bl * 256 + tid);

    lds_copy_wait();
    __syncthreads();

    const float sm[4] = {smu.x,  smu.y,  smu.z,  smu.w};
    const float sv[4] = {svar.x, svar.y, svar.z, svar.w};
    float mo[4], vo[4];
#pragma unroll
    for (int m = 0; m < 4; ++m) {
        const int idx = i * 32 + j4 + m;
        const float v1   = dexp2f(sv[m] * F_2LOG2E);   // exp(2*s_var)
        const float v2   = s_v2c[idx];                 // exp(2*tc_var)
        const float vadd = v1 + v2;
        const float r    = drcpf(vadd);
        const float lv   = dlog2f(vadd) * F_LN2;
        mo[m] = (sm[m] * v2 + s_tcmu[idx] * v1) * r;
        vo[m] = sv[m] + s_tcvar[idx] - 0.5f * lv;
    }
    v4f mu4 = {mo[0], mo[1], mo[2], mo[3]};
    v4f va4 = {vo[0], vo[1], vo[2], vo[3]};
    ((v4f*)cs_mu )[(size_t)bl * 256 + tid] = mu4;   // RT: re-read by kernel B (L2-resident)
    ((v4f*)cs_var)[(size_t)bl * 256 + tid] = va4;
}

// ---------------------------------------------------------------------------
// Kernel B: csp_scale[b,l,i,j,k] = overlap-scale(cs[b,l,i,j], trans_p[j,k])
// Block: one (b,l) and IPB consecutive i's; thread t: j = t>>3, k4 = (t&7)*4.
// 266 MB streaming output via NT v4f stores, perfectly coalesced over k.
// ---------------------------------------------------------------------------
#define IPB 8

extern "C" __global__ void __launch_bounds__(256)
lveg_csp_kernel(const float* __restrict__ tp_mu, const float* __restrict__ tp_var,
                const float* __restrict__ cs_mu, const float* __restrict__ cs_var,
                float* __restrict__ csp)
{
    __shared__ __align__(16) float s_tpmu[1024];
    __shared__ __align__(16) float s_v2p[1024];   // exp(2*tp_var)

    const int tid = threadIdx.x;
    const int bid = blockIdx.x;          // (b*127 + l)*4 + itile   (32/IPB == 4)
    const int bl  = bid >> 2;            // b*127 + l
    const int i0  = (bid & 3) * IPB;
    const int b   = bl / 127;
    const int l   = bl - b * 127;
    const int j   = tid >> 3;
    const int k4  = (tid & 7) << 2;

    lds_copy16(tp_mu + tid * 4, s_tpmu + tid * 4);

    v4f pv = ((const v4f*)tp_var)[tid];
    v4f w;
    w.x = dexp2f(pv.x * F_2LOG2E);
    w.y = dexp2f(pv.y * F_2LOG2E);
    w.z = dexp2f(pv.z * F_2LOG2E);
    w.w = dexp2f(pv.w * F_2LOG2E);
    ((v4f*)s_v2p)[tid] = w;

    const int csbase = ((b * 128 + l) * 32) * 32 + j;   // + i*32 below (LEN stride = 128)

    lds_copy_wait();
    __syncthreads();

    float w2[4], pm[4];
#pragma unroll
    for (int m = 0; m < 4; ++m) {
        w2[m] = s_v2p[j * 32 + k4 + m];
        pm[m] = s_tpmu[j * 32 + k4 + m];
    }

#pragma unroll
    for (int ii = 0; ii < IPB; ++ii) {
        const int   i   = i0 + ii;
        const float mu1 = cs_mu[csbase + i * 32];
        const float v1  = dexp2f(cs_var[csbase + i * 32] * F_2LOG2E);  // exp(2*cs_var)
        float o[4];
#pragma unroll
        for (int m = 0; m < 4; ++m) {
            const float vadd = v1 + w2[m];
            const float lv   = dlog2f(vadd) * F_LN2;
            const float d    = mu1 - pm[m];
            o[m] = -0.5f * (F_LOG2PI + lv + d * d * drcpf(vadd));
        }
        v4f o4 = {o[0], o[1], o[2], o[3]};
        __builtin_nontemporal_store(o4, (v4f*)csp + ((size_t)(bl * 32 + i) << 8) + tid);
    }
}

// ---------------------------------------------------------------------------
// Launch
// ---------------------------------------------------------------------------
extern "C" void kernel_launch(void* const* d_in, const int* in_sizes, int n_in,
                              void* d_out, int out_size, void* d_ws, size_t ws_size,
                              hipStream_t stream)
{
    (void)in_sizes; (void)n_in; (void)out_size; (void)d_ws; (void)ws_size;

    const int*   tokens  = (const int*)  d_in[0];
    const float* tp_mu   = (const float*)d_in[1];
    const float* tp_var  = (const float*)d_in[2];
    const float* tc_mu   = (const float*)d_in[3];
    const float* tc_var  = (const float*)d_in[4];
    const float* s_mu_t  = (const float*)d_in[5];
    const float* s_var_t = (const float*)d_in[6];

    float* out = (float*)d_out;
    const size_t n_csp = (size_t)16 * 127 * 32 * 32 * 32;  // 66,584,576
    const size_t n_cs  = (size_t)16 * 128 * 32 * 32;       //  2,097,152
    float* csp     = out;
    float* cs_mu   = out + n_csp;
    float* cs_var  = cs_mu  + n_cs;
    float* tpmu_e  = cs_var + n_cs;
    float* tpvar_e = tpmu_e + n_cs;

    lveg_cs_kernel<<<dim3(16 * 128), dim3(256), 0, stream>>>(
        tokens, tp_mu, tp_var, tc_mu, tc_var, s_mu_t, s_var_t,
        cs_mu, cs_var, tpmu_e, tpvar_e);

    lveg_csp_kernel<<<dim3(16 * 127 * 4), dim3(256), 0, stream>>>(
        tp_mu, tp_var, cs_mu, cs_var, csp);
}